// DeriveLayer_31903017075278
// MI455X (gfx1250) — compile-verified
//
#include <hip/hip_runtime.h>
#include <math.h>

// Problem constants (from the reference)
#define NLEN    16384            // input row length
#define WRES    16372            // width after trim(4) + 9-tap s1 pool
#define W2      16364            // stride-1 second-pool width
#define W9      1819             // stride-9 second-pool width  ((WRES-9)/9+1)
#define THREADS 256

#if __has_builtin(__builtin_amdgcn_global_load_async_to_lds_b128)
#define HAVE_ASYNC_LDS 1
#else
#define HAVE_ASYNC_LDS 0
#endif

// Signature (from hipcc diagnostic): (v4i addrspace(1)*, v4i addrspace(3)*, imm offset, imm cpol)
typedef int v4i __attribute__((ext_vector_type(4)));
typedef __attribute__((address_space(1))) v4i* gptr_v4i;
typedef __attribute__((address_space(3))) v4i* lptr_v4i;

__global__ __launch_bounds__(THREADS)
void derive_pool_kernel(const float* __restrict__ x, float* __restrict__ out)
{
    __shared__ __align__(16) float s_x[NLEN];        // 64 KB: staged input row
    __shared__ __align__(16) float s_res[WRES + 4];  // 64 KB: res row (padded)

    const int b = (int)blockIdx.x >> 2;   // batch row
    const int c = (int)blockIdx.x & 3;    // derivative channel (uniform per block)
    const float* __restrict__ xrow = x + (size_t)b * NLEN;

    // ---------- Stage 1: x row -> LDS (async global->LDS path, tracked by ASYNCcnt) ----------
#if HAVE_ASYNC_LDS
    {
        const int t = (int)threadIdx.x;
        #pragma unroll
        for (int it = 0; it < NLEN / (THREADS * 4); ++it) {
            const int idx = (it * THREADS + t) * 4;   // 16B-aligned chunks
            __builtin_amdgcn_global_load_async_to_lds_b128(
                (gptr_v4i)(xrow + idx),
                (lptr_v4i)(&s_x[idx]),
                /*offset=*/0, /*cpol=*/0);
        }
    }
  #if __has_builtin(__builtin_amdgcn_s_wait_asynccnt)
    __builtin_amdgcn_s_wait_asynccnt(0);
  #else
    asm volatile("s_wait_asynccnt 0" ::: "memory");
  #endif
#else
    for (int idx = (int)threadIdx.x * 4; idx < NLEN; idx += THREADS * 4) {
        *(float4*)(&s_x[idx]) = *(const float4*)(xrow + idx);
    }
#endif
    __syncthreads();

    // ---------- Stage 2: res[i] via telescoped 9-tap avg of shift-differences ----------
    // res[c][i] = (1/9) * sum_{t=4+i}^{12+i} r_c[t], all indices in-range (no wrap survives trim)
    const float inv9 = 1.0f / 9.0f;
    for (int i = (int)threadIdx.x; i < WRES; i += THREADS) {
        float v;
        if (c == 0) {            // x - roll(x,1): telescopes
            v = s_x[i + 12] - s_x[i + 3];
        } else if (c == 1) {     // x - roll(x,2)
            v = (s_x[i + 12] + s_x[i + 11]) - (s_x[i + 3] + s_x[i + 2]);
        } else if (c == 2) {     // x - roll(x,4)
            v = (s_x[i + 12] + s_x[i + 11] + s_x[i + 10] + s_x[i + 9])
              - (s_x[i + 3]  + s_x[i + 2]  + s_x[i + 1]  + s_x[i + 0]);
        } else {                 // second derivative: r0 - roll(r0,1) telescopes to r0[12+i]-r0[3+i]
            v = (s_x[i + 12] - s_x[i + 11]) - (s_x[i + 3] - s_x[i + 2]);
        }
        s_res[i] = v * inv9;
    }
    __syncthreads();

    // ---------- Stage 3: four branches (avg/max x stride1/stride9, upsampled) + res ----------
    float* __restrict__ o = out + ((size_t)b * 16 + (size_t)c) * WRES;
    const size_t bstride = (size_t)4 * WRES;   // branch stride in channels

    for (int i = (int)threadIdx.x; i < WRES; i += THREADS) {
        const float r = s_res[i];

        // stride-1 second pool, nearest-upsample index: j = i*W2/WRES (const-div -> mul/shift)
        const int base1 = (int)(((unsigned)i * (unsigned)W2) / (unsigned)WRES);
        float s1 = 0.0f, m1 = -INFINITY;
        #pragma unroll
        for (int j = 0; j < 9; ++j) {
            const float v = s_res[base1 + j];
            s1 += v;
            m1 = fmaxf(m1, v);
        }

        // stride-9 second pool: k = i*W9/WRES, window base 9k
        const int base9 = 9 * (int)(((unsigned)i * (unsigned)W9) / (unsigned)WRES);
        float s2 = 0.0f, m2 = -INFINITY;
        #pragma unroll
        for (int j = 0; j < 9; ++j) {
            const float v = s_res[base9 + j];
            s2 += v;
            m2 = fmaxf(m2, v);
        }

        o[i]                 = s1 * inv9 + r;  // branch 0: avg, stride 1
        o[bstride + i]       = s2 * inv9 + r;  // branch 1: avg, stride 9
        o[2 * bstride + i]   = m1 + r;         // branch 2: max, stride 1
        o[3 * bstride + i]   = m2 + r;         // branch 3: max, stride 9
    }
}

extern "C" void kernel_launch(void* const* d_in, const int* in_sizes, int n_in,
                              void* d_out, int out_size, void* d_ws, size_t ws_size,
                              hipStream_t stream)
{
    const float* x  = (const float*)d_in[0];
    float*      out = (float*)d_out;
    const int batch = in_sizes[0] / NLEN;   // 256
    dim3 grid((unsigned)(batch * 4)), block(THREADS);
    hipLaunchKernelGGL(derive_pool_kernel, grid, block, 0, stream, x, out);
}